// Mamba_model_28278064677588
// MI455X (gfx1250) — compile-verified
//
#include <hip/hip_runtime.h>
#include <hip/hip_bf16.h>
#include <math.h>

// ---------------------------------------------------------------------------
// Mamba forward for MI455X (gfx1250, wave32).
// GEMMs use V_WMMA_F32_16X16X4_F32 (fp32 matrix path, exact fp32 accumulate).
// Each wave owns a 32x64 output tile (2 M-tiles x NT N-tiles), K-loop is
// fully branch-free (tile count + epilogue are template params).
// Selective scan: one wave = 2 channels x 16 states, software-pipelined
// loads, cross-lane shfl_xor reduction for einsum('bdn,bn->bd').
// ---------------------------------------------------------------------------

#define D_MODEL  768
#define D_STATE  16
#define D_CONV   4
#define D_INNER  1536
#define DT_RANK  48
#define N_LAYERS 4
#define N_VOCAB  2048
#define LN_EPS   1e-5f
#define BATCH    2
#define SEQLEN   2048
#define ROWS     (BATCH * SEQLEN)          // 4096
#define DBL_W    (DT_RANK + 2 * D_STATE)   // 80

typedef __attribute__((ext_vector_type(2))) float v2f;
typedef __attribute__((ext_vector_type(8))) float v8f;

__device__ __forceinline__ float silu_f(float x) { return x / (1.0f + __expf(-x)); }
__device__ __forceinline__ float softplus_f(float x) {
    return (x > 20.0f) ? x : log1pf(__expf(x));
}

// ---------------------------------------------------------------------------
// fp32 WMMA GEMM: C[M,N] = A[M,K(lda)] * B[K,N(ldb)] (+ epilogue)
// Block = 256 threads = 8 waves; wave w owns rows [by*256 + w*32, +32),
// cols [bx*NT*16, +NT*16).
//
// fp32 WMMA VGPR layout (ISA 7.12.2):
//   A 16x4 : lanes 0-15 -> M=lane, v.x=K0,v.y=K1 ; lanes 16-31 -> K2,K3
//   B 4x16 : lanes 0-15 -> N=lane, v.x=row K0, v.y=row K1 ; lanes 16-31 -> K2,K3
//   C 16x16: VGPR j: lanes 0-15 = (M=j, N=lane), lanes 16-31 = (M=j+8, N=lane-16)
//
// MODE: 0 = none, 1 = softplus(acc + bias[col]), 2 = acc + res[row,col],
//       3 = acc + bias[col]
// ---------------------------------------------------------------------------
template <int NT, int MODE>
__global__ __launch_bounds__(256) void gemm_f32_wmma(
    const float* __restrict__ A, int lda,
    const float* __restrict__ B, int ldb,
    float* __restrict__ C, int ldc,
    const float* __restrict__ bias,
    const float* __restrict__ res,
    int K)
{
    const int wave = threadIdx.x >> 5;
    const int lane = threadIdx.x & 31;
    const int half = lane >> 4;   // 0: K0/K1 side, 1: K2/K3 side
    const int l16  = lane & 15;

    const int m0 = blockIdx.y * 256 + wave * 32;
    const int n0 = blockIdx.x * (NT * 16);

    v8f acc[2][NT];
#pragma unroll
    for (int mt = 0; mt < 2; ++mt)
#pragma unroll
        for (int t = 0; t < NT; ++t)
            acc[mt][t] = (v8f){0.f, 0.f, 0.f, 0.f, 0.f, 0.f, 0.f, 0.f};

    // Lane-resident base pointers (all lda multiples of 16 -> rows 64B aligned,
    // k + 2*half is even -> v2f loads are 8B aligned).
    const float* arow0 = A + (size_t)(m0 + l16) * lda + 2 * half;
    const float* arow1 = arow0 + (size_t)16 * lda;
    const float* bcol  = B + (size_t)(2 * half) * ldb + (n0 + l16);

#pragma unroll 2
    for (int k = 0; k < K; k += 4) {
        const v2f a0 = *(const v2f*)(arow0 + k);
        const v2f a1 = *(const v2f*)(arow1 + k);
        v2f bf[NT];
#pragma unroll
        for (int t = 0; t < NT; ++t) {
            const float* bp = bcol + (size_t)k * ldb + t * 16;
            bf[t].x = bp[0];
            bf[t].y = bp[ldb];
        }
#pragma unroll
        for (int t = 0; t < NT; ++t) {
            acc[0][t] = __builtin_amdgcn_wmma_f32_16x16x4_f32(
                false, a0, false, bf[t], (short)0, acc[0][t], false, false);
            acc[1][t] = __builtin_amdgcn_wmma_f32_16x16x4_f32(
                false, a1, false, bf[t], (short)0, acc[1][t], false, false);
        }
    }

#pragma unroll
    for (int mt = 0; mt < 2; ++mt)
#pragma unroll
        for (int t = 0; t < NT; ++t) {
            const int col = n0 + t * 16 + l16;
#pragma unroll
            for (int j = 0; j < 8; ++j) {
                const int row = m0 + mt * 16 + j + 8 * half;
                float v = acc[mt][t][j];
                if (MODE == 1)       v = softplus_f(v + bias[col]);
                else if (MODE == 2)  v += res[(size_t)row * ldc + col];
                else if (MODE == 3)  v += bias[col];
                C[(size_t)row * ldc + col] = v;
            }
        }
}

// ---------------------------------------------------------------------------
// Embedding gather: x[row, col] = emb[tokens[row], col]
// ---------------------------------------------------------------------------
__global__ __launch_bounds__(256) void embed_kernel(
    const int* __restrict__ tokens, const float* __restrict__ emb,
    float* __restrict__ x)
{
    int idx = blockIdx.x * blockDim.x + threadIdx.x;
    if (idx >= ROWS * D_MODEL) return;
    int row = idx / D_MODEL;
    int col = idx - row * D_MODEL;
    x[idx] = emb[(size_t)tokens[row] * D_MODEL + col];
}

// ---------------------------------------------------------------------------
// Depthwise causal conv (width 4, left pad 3) + bias + SiLU.
// ---------------------------------------------------------------------------
__global__ __launch_bounds__(256) void conv_silu_kernel(
    const float* __restrict__ xz, const float* __restrict__ conv_w,
    const float* __restrict__ conv_b, float* __restrict__ u)
{
    int idx = blockIdx.x * blockDim.x + threadIdx.x;
    if (idx >= ROWS * D_INNER) return;
    int c = idx % D_INNER;
    int l = (idx / D_INNER) % SEQLEN;
    int b = idx / (D_INNER * SEQLEN);

    const float* w = conv_w + (size_t)c * D_CONV;
    float s = conv_b[c];
#pragma unroll
    for (int j = 0; j < D_CONV; ++j) {
        int t = l - (D_CONV - 1) + j;
        if (t >= 0)
            s += w[j] * xz[((size_t)b * SEQLEN + t) * (2 * D_INNER) + c];
    }
    u[idx] = silu_f(s);
}

// ---------------------------------------------------------------------------
// Selective scan, software-pipelined: next timestep's operands are loaded
// before the current step's exp/fma/reduce chain executes.
// One wave handles 2 channels: lane = (d_half<<4) | n.
// ---------------------------------------------------------------------------
__global__ __launch_bounds__(256) void scan_kernel(
    const float* __restrict__ delta, const float* __restrict__ u,
    const float* __restrict__ dbl, const float* __restrict__ xz,
    const float* __restrict__ A_log, const float* __restrict__ D_skip,
    float* __restrict__ ybuf)
{
    int wid  = blockIdx.x * (blockDim.x >> 5) + (threadIdx.x >> 5);
    int lane = threadIdx.x & 31;
    int n    = lane & 15;
    int b    = wid / (D_INNER / 2);
    int d    = (wid % (D_INNER / 2)) * 2 + (lane >> 4);
    if (b >= BATCH) return;

    const float A   = -__expf(A_log[(size_t)d * D_STATE + n]);
    const float Dsk = D_skip[d];

    const size_t row0 = (size_t)b * SEQLEN;
    float dt = delta[row0 * D_INNER + d];
    float ut = u[row0 * D_INNER + d];
    float Bn = dbl[row0 * DBL_W + DT_RANK + n];
    float Cn = dbl[row0 * DBL_W + DT_RANK + D_STATE + n];

    float h = 0.0f;
    for (int t = 0; t < SEQLEN; ++t) {
        const size_t row = row0 + t;

        // Prefetch next timestep while this step's dependency chain runs.
        float dt1 = 0.f, ut1 = 0.f, Bn1 = 0.f, Cn1 = 0.f;
        if (t + 1 < SEQLEN) {
            const size_t rn = row + 1;
            dt1 = delta[rn * D_INNER + d];
            ut1 = u[rn * D_INNER + d];
            Bn1 = dbl[rn * DBL_W + DT_RANK + n];
            Cn1 = dbl[rn * DBL_W + DT_RANK + D_STATE + n];
        }

        const float dA = __expf(dt * A);
        h = dA * h + (dt * ut) * Bn;

        float p = h * Cn;
        p += __shfl_xor(p, 1, 32);   // reduce within the 16-lane half
        p += __shfl_xor(p, 2, 32);
        p += __shfl_xor(p, 4, 32);
        p += __shfl_xor(p, 8, 32);

        if (n == 0) {
            const float zt = xz[row * (2 * D_INNER) + D_INNER + d];
            ybuf[row * D_INNER + d] = (p + ut * Dsk) * silu_f(zt);
        }

        dt = dt1; ut = ut1; Bn = Bn1; Cn = Cn1;
    }
}

// ---------------------------------------------------------------------------
// LayerNorm over D_MODEL per row (block-per-row, shared tree reduce).
// ---------------------------------------------------------------------------
__global__ __launch_bounds__(256) void ln_kernel(
    const float* __restrict__ in, const float* __restrict__ g,
    const float* __restrict__ bta, float* __restrict__ out)
{
    __shared__ float sbuf[256];
    const int row = blockIdx.x;
    const float* r = in + (size_t)row * D_MODEL;

    float s = 0.0f;
    for (int i = threadIdx.x; i < D_MODEL; i += 256) s += r[i];
    sbuf[threadIdx.x] = s;
    __syncthreads();
    for (int off = 128; off > 0; off >>= 1) {
        if (threadIdx.x < off) sbuf[threadIdx.x] += sbuf[threadIdx.x + off];
        __syncthreads();
    }
    const float mu = sbuf[0] * (1.0f / D_MODEL);
    __syncthreads();

    float v = 0.0f;
    for (int i = threadIdx.x; i < D_MODEL; i += 256) {
        float dd = r[i] - mu;
        v += dd * dd;
    }
    sbuf[threadIdx.x] = v;
    __syncthreads();
    for (int off = 128; off > 0; off >>= 1) {
        if (threadIdx.x < off) sbuf[threadIdx.x] += sbuf[threadIdx.x + off];
        __syncthreads();
    }
    const float rstd = rsqrtf(sbuf[0] * (1.0f / D_MODEL) + LN_EPS);

    for (int i = threadIdx.x; i < D_MODEL; i += 256)
        out[(size_t)row * D_MODEL + i] = (r[i] - mu) * rstd * g[i] + bta[i];
}

// ---------------------------------------------------------------------------
extern "C" void kernel_launch(void* const* d_in, const int* in_sizes, int n_in,
                              void* d_out, int out_size, void* d_ws, size_t ws_size,
                              hipStream_t stream)
{
    const int*   tokens     = (const int*)  d_in[0];
    const float* emb        = (const float*)d_in[1];
    const float* in_proj_w  = (const float*)d_in[2];   // [4, 768, 3072]
    const float* conv_w     = (const float*)d_in[3];   // [4, 1536, 4]
    const float* conv_b     = (const float*)d_in[4];   // [4, 1536]
    const float* x_proj_w   = (const float*)d_in[5];   // [4, 1536, 80]
    const float* dt_proj_w  = (const float*)d_in[6];   // [4, 48, 1536]
    const float* dt_proj_b  = (const float*)d_in[7];   // [4, 1536]
    const float* A_log      = (const float*)d_in[8];   // [4, 1536, 16]
    const float* skip_D     = (const float*)d_in[9];   // [4, 1536]
    const float* out_proj_w = (const float*)d_in[10];  // [4, 1536, 768]
    const float* ln_g       = (const float*)d_in[11];  // [4, 768]
    const float* ln_b       = (const float*)d_in[12];  // [4, 768]
    const float* head_w     = (const float*)d_in[13];  // [768, 2048]
    const float* head_b     = (const float*)d_in[14];  // [2048]

    // Workspace layout (floats), ~152.5 MB total.
    float* ws   = (float*)d_ws;
    size_t off  = 0;
    float* x    = ws + off; off += (size_t)ROWS * D_MODEL;       // 4096 x 768
    float* xz   = ws + off; off += (size_t)ROWS * 2 * D_INNER;   // 4096 x 3072
    float* u    = ws + off; off += (size_t)ROWS * D_INNER;       // 4096 x 1536
    float* dbl  = ws + off; off += (size_t)ROWS * DBL_W;         // 4096 x 80
    float* dlt  = ws + off; off += (size_t)ROWS * D_INNER;       // 4096 x 1536
    float* ybuf = ws + off; off += (size_t)ROWS * D_INNER;       // 4096 x 1536
    float* tmp  = ws + off; off += (size_t)ROWS * D_MODEL;       // 4096 x 768

    const dim3 blk(256);
    const int  gy = ROWS / 256;   // 16 (wave tile = 32 rows, 8 waves/block)

    embed_kernel<<<(ROWS * D_MODEL + 255) / 256, blk, 0, stream>>>(tokens, emb, x);

    for (int layer = 0; layer < N_LAYERS; ++layer) {
        const float* W_in  = in_proj_w  + (size_t)layer * D_MODEL * 2 * D_INNER;
        const float* cw    = conv_w     + (size_t)layer * D_INNER * D_CONV;
        const float* cb    = conv_b     + (size_t)layer * D_INNER;
        const float* W_x   = x_proj_w   + (size_t)layer * D_INNER * DBL_W;
        const float* W_dt  = dt_proj_w  + (size_t)layer * DT_RANK * D_INNER;
        const float* b_dt  = dt_proj_b  + (size_t)layer * D_INNER;
        const float* Al    = A_log      + (size_t)layer * D_INNER * D_STATE;
        const float* Dl    = skip_D     + (size_t)layer * D_INNER;
        const float* W_out = out_proj_w + (size_t)layer * D_INNER * D_MODEL;
        const float* gl    = ln_g       + (size_t)layer * D_MODEL;
        const float* bl    = ln_b       + (size_t)layer * D_MODEL;

        // xz = x @ W_in                       [4096,768]x[768,3072]
        gemm_f32_wmma<4, 0><<<dim3((2 * D_INNER) / 64, gy), blk, 0, stream>>>(
            x, D_MODEL, W_in, 2 * D_INNER, xz, 2 * D_INNER,
            nullptr, nullptr, D_MODEL);

        // u = silu(depthwise_conv(xc) + conv_b)
        conv_silu_kernel<<<(ROWS * D_INNER + 255) / 256, blk, 0, stream>>>(
            xz, cw, cb, u);

        // dbl = u @ W_x                       [4096,1536]x[1536,80]  (N=80: NT=1)
        gemm_f32_wmma<1, 0><<<dim3(DBL_W / 16, gy), blk, 0, stream>>>(
            u, D_INNER, W_x, DBL_W, dbl, DBL_W,
            nullptr, nullptr, D_INNER);

        // delta = softplus(dbl[:, :48] @ W_dt + b_dt)
        gemm_f32_wmma<4, 1><<<dim3(D_INNER / 64, gy), blk, 0, stream>>>(
            dbl, DBL_W, W_dt, D_INNER, dlt, D_INNER,
            b_dt, nullptr, DT_RANK);

        // selective scan + skip + SiLU gate -> ybuf
        scan_kernel<<<(BATCH * (D_INNER / 2)) / 8, blk, 0, stream>>>(
            dlt, u, dbl, xz, Al, Dl, ybuf);

        // tmp = x + ybuf @ W_out              [4096,1536]x[1536,768]
        gemm_f32_wmma<4, 2><<<dim3(D_MODEL / 64, gy), blk, 0, stream>>>(
            ybuf, D_INNER, W_out, D_MODEL, tmp, D_MODEL,
            nullptr, x, D_INNER);

        // x = layer_norm(tmp)
        ln_kernel<<<ROWS, blk, 0, stream>>>(tmp, gl, bl, x);
    }

    // out = x @ head_w + head_b               [4096,768]x[768,2048]
    gemm_f32_wmma<4, 3><<<dim3(N_VOCAB / 64, gy), blk, 0, stream>>>(
        x, D_MODEL, head_w, N_VOCAB, (float*)d_out, N_VOCAB,
        head_b, nullptr, D_MODEL);
}